// miniNN_29944511988290
// MI455X (gfx1250) — compile-verified
//
#include <hip/hip_runtime.h>

// ---------------------------------------------------------------------------
// out[p] = max(0, segment_max over idx of (feats[n,:32] . W + b))
//
// Memory-bound: 512MB feats stream -> ~23us floor @ 23.3 TB/s.
// Dot products on the matrix pipe via V_WMMA_F32_16X16X4_F32:
//   - one 16-row tile per WMMA group, K=32 as 8 accumulating K=4 WMMAs
//   - B = W broadcast to all 16 columns (same per-lane float2 pattern as A)
//   - each wave processes 4 consecutive tiles (8KB contiguous) with W/bias
//     hoisted to registers, so all steady-state VMEM is useful feats traffic
// Reduction via global_atomic_max_u32 on float bits (exact for positives;
// out initialized to +0.0f, r<=0 filtered before the atomic).
// ---------------------------------------------------------------------------

typedef float v2f __attribute__((ext_vector_type(2)));
typedef float v8f __attribute__((ext_vector_type(8)));

#define TILES_PER_WAVE 4   // 64 rows / wave

__global__ void seg_init_kernel(float* __restrict__ out, int p) {
    int i = blockIdx.x * blockDim.x + threadIdx.x;
    if (i < p) out[i] = 0.0f;
}

__global__ __launch_bounds__(256) void seg_linear_max_wmma(
    const float* __restrict__ feats,   // [N,32] row-major
    const float* __restrict__ Wp,      // [32]
    const float* __restrict__ bp,      // [1]
    const int*   __restrict__ idx,     // [N]
    float*       __restrict__ out,     // [P], pre-zeroed
    int ntiles)                        // N/16
{
    const int wave  = threadIdx.x >> 5;          // wave32
    const int lane  = threadIdx.x & 31;
    const int gwave = blockIdx.x * (blockDim.x >> 5) + wave;
    const int tile0 = gwave * TILES_PER_WAVE;
    if (tile0 >= ntiles) return;                 // wave-uniform

    const int m     = lane & 15;                 // row within a tile
    const int khalf = lane >> 4;                 // 0: K%4 in {0,1}; 1: {2,3}

    // Hoist W (B-matrix operand pairs) and bias: loaded once per wave.
    // B-matrix (4x16 f32) with every column = W needs, per lane,
    // (w[4c+2*khalf], w[4c+2*khalf+1]) -- identical pattern to the A loads.
    const float* wp = Wp + 2 * khalf;
    v2f wv[8];
#pragma unroll
    for (int c = 0; c < 8; ++c) wv[c] = *(const v2f*)(wp + 4 * c);
    const float bias = bp[0];

    const int tend = (tile0 + TILES_PER_WAVE < ntiles) ? tile0 + TILES_PER_WAVE
                                                       : ntiles;

    for (int tile = tile0; tile < tend; ++tile) {   // wave-uniform loop
        const int rb = tile * 16;

        // A-matrix (16x4 f32) layout: V0 = K0 (lanes 0-15) / K2 (lanes 16-31),
        // V1 = K1 / K3. Lane loads its row's (4c+2*khalf, +1) float2; a wave's
        // 8 loads cover rows rb..rb+15 * 128B = 2KB contiguous -> dense stream.
        const float* rowp = feats + (size_t)(rb + m) * 32 + 2 * khalf;

        v8f acc = {0.f, 0.f, 0.f, 0.f, 0.f, 0.f, 0.f, 0.f};
#pragma unroll
        for (int c = 0; c < 8; ++c) {
            v2f a = __builtin_nontemporal_load((const v2f*)(rowp + 4 * c));
            // (neg_a, A, neg_b, B, c_mod, C, reuse_a, reuse_b)
            acc = __builtin_amdgcn_wmma_f32_16x16x4_f32(
                false, a, false, wv[c], (short)0, acc, false, false);
        }

        // D layout: VGPR v = row rb+v (lanes 0-15) / rb+8+v (lanes 16-31);
        // all 16 columns identical. Lanes 0-7 emit rows rb..rb+7,
        // lanes 16-23 emit rows rb+8..rb+15.
        const bool active = (lane < 8) || (lane >= 16 && lane < 24);
        if (active) {
            const int j = lane & 7;
            float v = acc[0];
            v = (j == 1) ? acc[1] : v;
            v = (j == 2) ? acc[2] : v;
            v = (j == 3) ? acc[3] : v;
            v = (j == 4) ? acc[4] : v;
            v = (j == 5) ? acc[5] : v;
            v = (j == 6) ? acc[6] : v;
            v = (j == 7) ? acc[7] : v;
            v += bias;
            if (v > 0.0f) {  // r<=0 can never beat the 0.0f init: skip atomic
                const int row = rb + j + ((lane >= 16) ? 8 : 0);
                const int seg = __builtin_nontemporal_load(idx + row);
                // positive floats are order-isomorphic to their u32 bits
                atomicMax(reinterpret_cast<unsigned int*>(out) + seg,
                          __float_as_uint(v));
            }
        }
    }
}

// Generic tail for N % 16 != 0 (unused for N = 4,000,000).
__global__ void seg_tail_kernel(const float* __restrict__ feats,
                                const float* __restrict__ Wp,
                                const float* __restrict__ bp,
                                const int*   __restrict__ idx,
                                float*       __restrict__ out,
                                int start, int n)
{
    int i = start + blockIdx.x * blockDim.x + threadIdx.x;
    if (i < n) {
        float r = bp[0];
#pragma unroll
        for (int k = 0; k < 32; ++k) r += feats[(size_t)i * 32 + k] * Wp[k];
        if (r > 0.0f)
            atomicMax(reinterpret_cast<unsigned int*>(out) + idx[i],
                      __float_as_uint(r));
    }
}

extern "C" void kernel_launch(void* const* d_in, const int* in_sizes, int n_in,
                              void* d_out, int out_size, void* d_ws, size_t ws_size,
                              hipStream_t stream) {
    (void)n_in; (void)d_ws; (void)ws_size;
    const float* feats = (const float*)d_in[0];   // [N,32]
    const float* W     = (const float*)d_in[1];   // [1,32]
    const float* b     = (const float*)d_in[2];   // [1]
    const int*   idx   = (const int*)d_in[3];     // [N] (int32 per harness)
    float*       out   = (float*)d_out;           // [P]

    const long long n      = (long long)in_sizes[0] / 32;  // rows
    const int       ntiles = (int)(n / 16);
    const int       rem    = (int)(n - (long long)ntiles * 16);

    // out must be re-initialized every call (harness poisons once, never restores)
    seg_init_kernel<<<(out_size + 255) / 256, 256, 0, stream>>>(out, out_size);

    if (ntiles > 0) {
        const int wavesPerBlock = 8;  // 256 threads = 8 wave32
        const int tilesPerBlock = wavesPerBlock * TILES_PER_WAVE;
        const int blocks = (ntiles + tilesPerBlock - 1) / tilesPerBlock;
        seg_linear_max_wmma<<<blocks, 256, 0, stream>>>(feats, W, b, idx, out, ntiles);
    }
    if (rem > 0) {
        seg_tail_kernel<<<1, 32, 0, stream>>>(feats, W, b, idx, out,
                                              ntiles * 16, (int)n);
    }
}